// GAT_75213467287964
// MI455X (gfx1250) — compile-verified
//
#include <hip/hip_runtime.h>
#include <hip/hip_bf16.h>

#define GAT_N      50000
#define GAT_E      800000
#define GAT_ETOT   850000   // E + N self loops
#define GAT_D      128
#define GAT_G      64
#define NEG_SLOPE  0.2f

typedef __attribute__((ext_vector_type(2))) float v2f;
typedef __attribute__((ext_vector_type(8))) float v8f;

// ---- monotone float <-> uint encoding for atomicMax on floats ----
__device__ __forceinline__ unsigned enc_f32(float f) {
    unsigned u = __float_as_uint(f);
    return (u & 0x80000000u) ? ~u : (u | 0x80000000u);
}
__device__ __forceinline__ float dec_f32(unsigned k) {
    unsigned u = (k & 0x80000000u) ? (k & 0x7FFFFFFFu) : ~k;
    return __uint_as_float(u);
}

// =====================================================================
// h = X @ W   (X: [Nrows,128] row-major, W: [128,128] row-major)
// grid.x = Nrows/16 row tiles; 8 waves/block, wave w -> column tile w.
// Each wave: 16x16 f32 tile via 32x V_WMMA_F32_16X16X4_F32.
// =====================================================================
__global__ void gat_gemm128_wmma(const float* __restrict__ X,
                                 const float* __restrict__ W,
                                 float* __restrict__ H) {
    const int wave = threadIdx.x >> 5;     // 0..7
    const int lane = threadIdx.x & 31;
    const int half = lane >> 4;            // 0: K pair {k,k+1}, 1: {k+2,k+3}
    const int l    = lane & 15;
    const int m0   = blockIdx.x * 16;
    const int n0   = wave * 16;

    const float* xrow = X + (size_t)(m0 + l) * GAT_D;
    v8f acc = {};

#pragma unroll
    for (int k = 0; k < GAT_D; k += 4) {
        const int kk = k + 2 * half;
        // A fragment: row m0+l, K = kk, kk+1  (8B-aligned pair)
        v2f a;
        {
            const float2 av = *(const float2*)(xrow + kk);
            a.x = av.x; a.y = av.y;
        }
        // B fragment: col n0+l, K rows kk, kk+1
        v2f b;
        b.x = W[(size_t)kk       * GAT_D + n0 + l];
        b.y = W[(size_t)(kk + 1) * GAT_D + n0 + l];
        acc = __builtin_amdgcn_wmma_f32_16x16x4_f32(
            false, a, false, b, (short)0, acc, false, false);
    }

    // C/D layout: VGPR r -> row (m0 + half*8 + r), col n0 + l
    const int mBase = m0 + half * 8;
#pragma unroll
    for (int r = 0; r < 8; ++r)
        H[(size_t)(mBase + r) * GAT_D + n0 + l] = acc[r];
}

// =====================================================================
// alpha_s[n] = h[n,:] . a_src ; alpha_d[n] = h[n,:] . a_dst
// one wave (32 lanes) per node, 8 nodes per 256-thread block
// =====================================================================
__global__ void gat_alpha(const float* __restrict__ H,
                          const float* __restrict__ a_src,
                          const float* __restrict__ a_dst,
                          float* __restrict__ as_out,
                          float* __restrict__ ad_out) {
    const int node = blockIdx.x * 8 + (threadIdx.x >> 5);
    const int lane = threadIdx.x & 31;
    if (node >= GAT_N) return;
    const float* h = H + (size_t)node * GAT_D;
    float s = 0.f, d = 0.f;
#pragma unroll
    for (int i = lane; i < GAT_D; i += 32) {
        const float v = h[i];
        s += v * a_src[i];
        d += v * a_dst[i];
    }
#pragma unroll
    for (int off = 16; off > 0; off >>= 1) {
        s += __shfl_xor(s, off, 32);
        d += __shfl_xor(d, off, 32);
    }
    if (lane == 0) { as_out[node] = s; ad_out[node] = d; }
}

// =====================================================================
// out[n,d] = bias[d];  m_u[n] = 0 (== -inf encoded-less-than-anything);
// ssum[n] = 0
// grid covers N*D threads
// =====================================================================
__global__ void gat_init(float* __restrict__ out,
                         const float* __restrict__ bias,
                         unsigned* __restrict__ m_u,
                         float* __restrict__ ssum) {
    const size_t t = (size_t)blockIdx.x * blockDim.x + threadIdx.x;
    if (t < (size_t)GAT_N * GAT_D) out[t] = bias[t & (GAT_D - 1)];
    if (t < GAT_N) { m_u[t] = 0u; ssum[t] = 0.f; }
}

// =====================================================================
// pass1: e = leakyrelu(alpha_s[src] + alpha_d[dst]); segment max via
// integer atomicMax on monotone encoding
// =====================================================================
__global__ void gat_edge_max(const int* __restrict__ ei,
                             const float* __restrict__ as,
                             const float* __restrict__ ad,
                             float* __restrict__ ebuf,
                             unsigned* __restrict__ m_u) {
    const int i = blockIdx.x * blockDim.x + threadIdx.x;
    if (i >= GAT_ETOT) return;
    int s_idx, d_idx;
    if (i < GAT_E) { s_idx = ei[i]; d_idx = ei[GAT_E + i]; }
    else           { s_idx = d_idx = i - GAT_E; }
    float e = as[s_idx] + ad[d_idx];
    e = (e > 0.f) ? e : NEG_SLOPE * e;
    ebuf[i] = e;
    atomicMax(&m_u[d_idx], enc_f32(e));
}

// =====================================================================
// pass2: ex = exp(e - m[dst]); segment sum of ex
// =====================================================================
__global__ void gat_edge_exp(const int* __restrict__ ei,
                             float* __restrict__ ebuf,
                             const unsigned* __restrict__ m_u,
                             float* __restrict__ ssum) {
    const int i = blockIdx.x * blockDim.x + threadIdx.x;
    if (i >= GAT_ETOT) return;
    int d_idx;
    if (i < GAT_E) d_idx = ei[GAT_E + i];
    else           d_idx = i - GAT_E;
    const float m  = dec_f32(m_u[d_idx]);
    const float ex = expf(ebuf[i] - m);
    ebuf[i] = ex;
    atomicAdd(&ssum[d_idx], ex);
}

// =====================================================================
// pass3: coef = ex / (s[dst]+1e-16); out[dst,:] += coef * h[src,:]
// one wave per edge (32 lanes x float4 = 128 floats), 8 edges/block
// =====================================================================
__global__ void gat_edge_scatter(const int* __restrict__ ei,
                                 const float* __restrict__ ebuf,
                                 const float* __restrict__ ssum,
                                 const float* __restrict__ H,
                                 float* __restrict__ out) {
    const int edge = blockIdx.x * 8 + (threadIdx.x >> 5);
    const int lane = threadIdx.x & 31;
    if (edge >= GAT_ETOT) return;
    int s_idx, d_idx;
    if (edge < GAT_E) { s_idx = ei[edge]; d_idx = ei[GAT_E + edge]; }
    else              { s_idx = d_idx = edge - GAT_E; }
    const float coef = ebuf[edge] / (ssum[d_idx] + 1e-16f);
    const float4 h4 = *(const float4*)(H + (size_t)s_idx * GAT_D + lane * 4);
    float* o = out + (size_t)d_idx * GAT_D + lane * 4;
    atomicAdd(o + 0, h4.x * coef);
    atomicAdd(o + 1, h4.y * coef);
    atomicAdd(o + 2, h4.z * coef);
    atomicAdd(o + 3, h4.w * coef);
}

// =====================================================================
// pooling
// =====================================================================
__global__ void gat_pool_zero(float* __restrict__ pool, unsigned* __restrict__ cnt) {
    const int t = blockIdx.x * blockDim.x + threadIdx.x;
    if (t < GAT_G * GAT_D) pool[t] = 0.f;
    if (t < GAT_G) cnt[t] = 0u;
}

__global__ void gat_pool_accum(const float* __restrict__ feats,
                               const int* __restrict__ batch,
                               float* __restrict__ pool,
                               unsigned* __restrict__ cnt) {
    const size_t t = (size_t)blockIdx.x * blockDim.x + threadIdx.x;
    if (t >= (size_t)GAT_N * GAT_D) return;
    const int n = (int)(t >> 7);
    const int d = (int)(t & (GAT_D - 1));
    const int g = batch[n];
    atomicAdd(&pool[g * GAT_D + d], feats[t]);
    if (d == 0) atomicAdd(&cnt[g], 1u);
}

__global__ void gat_pool_final(const float* __restrict__ pool,
                               const unsigned* __restrict__ cnt,
                               float* __restrict__ out_pooled) {
    const int t = blockIdx.x * blockDim.x + threadIdx.x;
    if (t >= GAT_G * GAT_D) return;
    const float c = fmaxf((float)cnt[t >> 7], 1.0f);
    out_pooled[t] = pool[t] / c;
}

// =====================================================================
// host side
// =====================================================================
static void launch_gat_layer(const float* x, const float* W,
                             const float* a_s, const float* a_d,
                             const float* bias,
                             const int* ei,
                             float* h, float* alphaS, float* alphaD,
                             unsigned* m_u, float* ssum, float* ebuf,
                             float* out, hipStream_t stream) {
    gat_gemm128_wmma<<<GAT_N / 16, 256, 0, stream>>>(x, W, h);
    gat_alpha<<<GAT_N / 8, 256, 0, stream>>>(h, a_s, a_d, alphaS, alphaD);
    gat_init<<<(GAT_N * GAT_D) / 256, 256, 0, stream>>>(out, bias, m_u, ssum);
    const int eb = (GAT_ETOT + 255) / 256;
    gat_edge_max<<<eb, 256, 0, stream>>>(ei, alphaS, alphaD, ebuf, m_u);
    gat_edge_exp<<<eb, 256, 0, stream>>>(ei, ebuf, m_u, ssum);
    gat_edge_scatter<<<GAT_ETOT / 8, 256, 0, stream>>>(ei, ebuf, ssum, h, out);
}

extern "C" void kernel_launch(void* const* d_in, const int* in_sizes, int n_in,
                              void* d_out, int out_size, void* d_ws, size_t ws_size,
                              hipStream_t stream) {
    const float* x     = (const float*)d_in[0];
    const int*   ei    = (const int*)d_in[1];   // [2,E] flat: src then dst
    const int*   batch = (const int*)d_in[2];
    const float* W1  = (const float*)d_in[3];
    const float* as1 = (const float*)d_in[4];
    const float* ad1 = (const float*)d_in[5];
    const float* b1  = (const float*)d_in[6];
    const float* W2  = (const float*)d_in[7];
    const float* as2 = (const float*)d_in[8];
    const float* ad2 = (const float*)d_in[9];
    const float* b2  = (const float*)d_in[10];

    float* out_feats  = (float*)d_out;                          // [N,128]
    float* out_pooled = out_feats + (size_t)GAT_N * GAT_D;      // [64,128]

    // workspace layout (32-bit words)
    float* ws = (float*)d_ws;
    size_t o = 0;
    float*    h      = ws + o; o += (size_t)GAT_N * GAT_D;   // gemm output (reused both layers)
    float*    feats1 = ws + o; o += (size_t)GAT_N * GAT_D;   // layer-1 output
    float*    alphaS = ws + o; o += GAT_N;
    float*    alphaD = ws + o; o += GAT_N;
    unsigned* m_u    = (unsigned*)(ws + o); o += GAT_N;
    float*    ssum   = ws + o; o += GAT_N;
    float*    ebuf   = ws + o; o += GAT_ETOT;
    float*    pool   = ws + o; o += GAT_G * GAT_D;
    unsigned* cnt    = (unsigned*)(ws + o); o += GAT_G;
    (void)ws_size; (void)in_sizes; (void)n_in; (void)out_size;

    // layer 1: x -> feats1
    launch_gat_layer(x, W1, as1, ad1, b1, ei,
                     h, alphaS, alphaD, m_u, ssum, ebuf, feats1, stream);
    // layer 2: feats1 -> d_out feats
    launch_gat_layer(feats1, W2, as2, ad2, b2, ei,
                     h, alphaS, alphaD, m_u, ssum, ebuf, out_feats, stream);

    // global mean pool
    gat_pool_zero<<<(GAT_G * GAT_D + 255) / 256, 256, 0, stream>>>(pool, cnt);
    gat_pool_accum<<<(GAT_N * GAT_D) / 256, 256, 0, stream>>>(out_feats, batch, pool, cnt);
    gat_pool_final<<<(GAT_G * GAT_D + 255) / 256, 256, 0, stream>>>(pool, cnt, out_pooled);
}